// BackbonePredictor_11682311045509
// MI455X (gfx1250) — compile-verified
//
#include <hip/hip_runtime.h>
#include <cstddef>

// ---------------------------------------------------------------------------
// Bipartite-GNN forward for MI455X (gfx1250), fp32 end-to-end.
// - Per-edge Linear (ff_W) hoisted through the segment-sum (linearity), so all
//   dense math is node-level V_WMMA_F32_16X16X4_F32.
// - Edge loop: gather + wave32-butterfly LayerNorm + atomic scatter-add
//   (memory-bound: ~1.6 GB total @ 23.3 TB/s HBM).
// - GEMM: templated on (N,K); one wave per 16x64 C strip (4 WMMA
//   accumulators); branchless clamped loads; pure pointer-increment
//   addressing in the K loop (no runtime-stride multiplies).
// ---------------------------------------------------------------------------

#define LN_EPS 1e-5f

typedef __attribute__((ext_vector_type(2))) float v2f;
typedef __attribute__((ext_vector_type(8))) float v8f;

// ---------------- generic fp32 WMMA GEMM: C = [relu](A[MxK] @ B[KxN] + bias) --
// One wave computes a 16x64 strip of C (4 16x16 WMMA tiles sharing one A frag).
// Loads are branchless: A rows and B cols are clamped in-range; garbage only
// reaches C rows/cols that the store guard drops (each C row depends only on
// its A row, each C col on its B col).
template <int N, int K>
__global__ __launch_bounds__(256) void gemm16_wmma_f32(
    const float* __restrict__ A, const float* __restrict__ B,
    const float* __restrict__ bias, float* __restrict__ C,
    int M, int ldc, int relu)
{
    constexpr int tilesNG = (N + 63) / 64;         // groups of 4 N-tiles (64 cols)
    int tilesM = (M + 15) >> 4;
    int wave = blockIdx.x * (blockDim.x >> 5) + (threadIdx.x >> 5);
    if (wave >= tilesM * tilesNG) return;          // wave-uniform exit
    int tm, tg;
    if (tilesNG == 1) { tm = wave; tg = 0; }
    else              { tm = wave / tilesNG; tg = wave % tilesNG; }

    int lane = threadIdx.x & 31;
    int half = lane >> 4;                          // 0: lanes 0-15, 1: lanes 16-31
    int l    = lane & 15;

    int rowA = tm * 16 + l;                        // A fragment row (lane-held M)
    if (rowA >= M) rowA = M - 1;                   // clamp: row never stored anyway
    // ISA 7.12.2 A layout: VGPR0={K0|K2}, VGPR1={K1|K3} -> lane reads (ka, ka+1)
    const float* Ap = A + (size_t)rowA * K + 2 * half;

    const float* bc[4];                            // per-tile B column pointers
#pragma unroll
    for (int t = 0; t < 4; ++t) {
        int c = tg * 64 + t * 16 + l;              // B/C fragment column
        if (c >= N) c = N - 1;                     // clamp: col never stored anyway
        bc[t] = B + (size_t)(2 * half) * N + c;
    }

    v8f acc[4] = {{}, {}, {}, {}};
    for (int k0 = 0; k0 < K; k0 += 4) {
        v2f a = *(const v2f*)Ap;                   // global_load_b64
        Ap += 4;
#pragma unroll
        for (int t = 0; t < 4; ++t) {
            v2f b;
            b.x = bc[t][0];                        // row ka   (immediate offset 0)
            b.y = bc[t][N];                        // row ka+1 (immediate offset 4*N)
            bc[t] += (size_t)4 * N;                // advance 4 K-rows
            acc[t] = __builtin_amdgcn_wmma_f32_16x16x4_f32(
                false, a, false, b, (short)0, acc[t], false, false);
        }
    }

#pragma unroll
    for (int t = 0; t < 4; ++t) {
        int c = tg * 64 + t * 16 + l;
        if (c < N) {
            float bv = (bias != nullptr) ? bias[c] : 0.0f;
#pragma unroll
            for (int v = 0; v < 8; ++v) {          // C layout: reg v -> row v + 8*half
                int row = tm * 16 + v + 8 * half;
                if (row < M) {
                    float x = acc[t][v] + bv;
                    if (relu) x = fmaxf(x, 0.0f);
                    C[(size_t)row * ldc + c] = x;
                }
            }
        }
    }
}

// ---------------- edge LayerNorm over a length-1 axis (general form) ----------
__global__ void edge_ln_kernel(const float* __restrict__ ef,
                               const float* __restrict__ g, const float* __restrict__ b,
                               float* __restrict__ efn, int E)
{
    int e = blockIdx.x * blockDim.x + threadIdx.x;
    if (e >= E) return;
    float x = ef[e];
    float mu = x;                 // mean over 1 element
    float var = 0.0f;
    efn[e] = (x - mu) * rsqrtf(var + LN_EPS) * g[0] + b[0];
}

// ---------------- degree counts (cached once per direction) -------------------
__global__ void deg_kernel(const int* __restrict__ idx, float* __restrict__ deg, int E)
{
    int e = blockIdx.x * blockDim.x + threadIdx.x;
    if (e < E) atomicAdd(deg + idx[e], 1.0f);
}

// ---------------- embedding front: h1 = relu(LN_K(x) @ W1 + b1), K in {4,6} ---
template <int KF>
__global__ void embed_front_kernel(const float* __restrict__ X,
                                   const float* __restrict__ lng, const float* __restrict__ lnb,
                                   const float* __restrict__ W1, const float* __restrict__ b1,
                                   float* __restrict__ H1, int Nn)
{
    int r = blockIdx.x * blockDim.x + threadIdx.x;
    if (r >= Nn) return;
    float x[KF];
    float mean = 0.0f;
#pragma unroll
    for (int k = 0; k < KF; ++k) { x[k] = X[(size_t)r * KF + k]; mean += x[k]; }
    mean *= (1.0f / KF);
    float var = 0.0f;
#pragma unroll
    for (int k = 0; k < KF; ++k) { float d = x[k] - mean; var += d * d; }
    var *= (1.0f / KF);
    float rinv = rsqrtf(var + LN_EPS);
#pragma unroll
    for (int k = 0; k < KF; ++k) x[k] = (x[k] - mean) * rinv * lng[k] + lnb[k];
    for (int n = 0; n < 64; ++n) {
        float acc = b1[n];
#pragma unroll
        for (int k = 0; k < KF; ++k) acc += x[k] * W1[k * 64 + n];
        H1[(size_t)r * 64 + n] = fmaxf(acc, 0.0f);
    }
}

// ---------------- per-edge message: gather, LN64 (wave butterfly), relu, scatter
// One wave32 per edge; 2 floats per lane (float2, coalesced 256B rows).
__global__ __launch_bounds__(256) void edge_msg_kernel(
    const float* __restrict__ Tsrc, const float* __restrict__ Ttgt,
    const float* __restrict__ efn,  const float* __restrict__ feW,
    const float* __restrict__ lng,  const float* __restrict__ lnb,
    const int* __restrict__ src, const int* __restrict__ tgt,
    float* __restrict__ S, int E)
{
    int e = blockIdx.x * (blockDim.x >> 5) + (threadIdx.x >> 5);
    if (e >= E) return;                            // wave-uniform
    int lane = threadIdx.x & 31;

    int s = src[e], t = tgt[e];
    const float2 a  = ((const float2*)(Tsrc + (size_t)s * 64))[lane];
    const float2 r  = ((const float2*)(Ttgt + (size_t)t * 64))[lane];
    const float2 w  = ((const float2*)feW)[lane];
    float  ef = efn[e];

    float2 m;
    m.x = a.x + r.x + ef * w.x;
    m.y = a.y + r.y + ef * w.y;

    float sum = m.x + m.y;
    float ssq = m.x * m.x + m.y * m.y;
#pragma unroll
    for (int o = 16; o > 0; o >>= 1) {
        sum += __shfl_xor(sum, o, 32);
        ssq += __shfl_xor(ssq, o, 32);
    }
    float mean = sum * (1.0f / 64.0f);
    float var  = ssq * (1.0f / 64.0f) - mean * mean;
    float rinv = rsqrtf(var + LN_EPS);

    const float2 gg = ((const float2*)lng)[lane];
    const float2 bb = ((const float2*)lnb)[lane];
    float yx = fmaxf((m.x - mean) * rinv * gg.x + bb.x, 0.0f);
    float yy = fmaxf((m.y - mean) * rinv * gg.y + bb.y, 0.0f);

    atomicAdd(S + (size_t)t * 64 + 2 * lane,     yx);
    atomicAdd(S + (size_t)t * 64 + 2 * lane + 1, yy);
}

// ---------------- post-aggregation: t = LN(agg + deg*ff_b); concat [t, right] -
__global__ void ln_concat_kernel(const float* __restrict__ agg, const float* __restrict__ deg,
                                 const float* __restrict__ ffb,
                                 const float* __restrict__ lng, const float* __restrict__ lnb,
                                 const float* __restrict__ right, float* __restrict__ C128, int Nn)
{
    int r = blockIdx.x * blockDim.x + threadIdx.x;
    if (r >= Nn) return;
    float d = deg[r];
    float v[64];
    float mean = 0.0f;
#pragma unroll
    for (int k = 0; k < 64; ++k) { v[k] = agg[(size_t)r * 64 + k] + d * ffb[k]; mean += v[k]; }
    mean *= (1.0f / 64.0f);
    float var = 0.0f;
#pragma unroll
    for (int k = 0; k < 64; ++k) { float t = v[k] - mean; var += t * t; }
    var *= (1.0f / 64.0f);
    float rinv = rsqrtf(var + LN_EPS);
#pragma unroll
    for (int k = 0; k < 64; ++k)
        C128[(size_t)r * 128 + k] = (v[k] - mean) * rinv * lng[k] + lnb[k];
#pragma unroll
    for (int k = 0; k < 64; ++k)
        C128[(size_t)r * 128 + 64 + k] = right[(size_t)r * 64 + k];
}

// ---------------- LiteralsMessage: y = relu(a*x + b*flip(x)), flip = row^1 ----
__global__ void literals_kernel(const float* __restrict__ X,
                                const float* __restrict__ a, const float* __restrict__ b,
                                float* __restrict__ Y, int Nn)
{
    int i = blockIdx.x * blockDim.x + threadIdx.x;
    if (i >= Nn * 64) return;
    int row = i >> 6, k = i & 63;
    float flip = X[(size_t)(row ^ 1) * 64 + k];
    Y[i] = fmaxf(a[k] * X[i] + b[k] * flip, 0.0f);
}

// ---------------- join_literals: J[p][2k+j] = X[2p+j][k] ----------------------
__global__ void join_kernel(const float* __restrict__ X, float* __restrict__ J, int P)
{
    int i = blockIdx.x * blockDim.x + threadIdx.x;
    if (i >= P * 128) return;
    int p = i >> 7, c = i & 127, k = c >> 1, j = c & 1;
    J[i] = X[(size_t)(2 * p + j) * 64 + k];
}

// ===========================================================================
// Host side
// ===========================================================================

struct ConvP {
    const float *feW, *ffW, *ffb, *ff_lnb, *ff_lng, *flW, *flb, *frW,
                *o1W, *o1b, *o2W, *o2b, *pc_lnb, *pc_lng;
};

// conv_params leaves in JAX sorted-key order:
// fe_W, ff_W, ff_b, ff_ln_b, ff_ln_g, fl_W, fl_b, fr_W, o1_W, o1_b, o2_W, o2_b, pc_ln_b, pc_ln_g
static const int kConvSz[14] = {64, 4096, 64, 64, 64, 4096, 64, 4096, 8192, 64, 4096, 64, 64, 64};

// 77 params leaves, JAX pytree (sorted dict keys; ASCII: 'W' < 'b' < 'l') order:
// c2v[0](14) c2v[1](14) cons_W1 cons_W2 cons_b1 cons_b2 cons_ln_b cons_ln_g
// edge_ln_b edge_ln_g ltl0_a ltl0_b ltl1_a ltl1_b out_W1 out_W2 out_b1
// v2c[0](14) v2c[1](14) var_W1 var_W2 var_b1 var_b2 var_ln_b var_ln_g
static void build_leaves(void* const* d_in, int n_in, const float* leaf[77])
{
    int sz[77]; int i = 0;
    for (int c = 0; c < 2; ++c) for (int j = 0; j < 14; ++j) sz[i++] = kConvSz[j]; // c2v
    sz[i++] = 256; sz[i++] = 4096; sz[i++] = 64; sz[i++] = 64; sz[i++] = 4; sz[i++] = 4; // cons_*
    sz[i++] = 1; sz[i++] = 1;                                                            // edge_ln_b/g
    sz[i++] = 64; sz[i++] = 64; sz[i++] = 64; sz[i++] = 64;                              // ltl
    sz[i++] = 16384; sz[i++] = 384; sz[i++] = 128;                                       // out_W1/W2/b1
    for (int c = 0; c < 2; ++c) for (int j = 0; j < 14; ++j) sz[i++] = kConvSz[j]; // v2c
    sz[i++] = 384; sz[i++] = 4096; sz[i++] = 64; sz[i++] = 64; sz[i++] = 6; sz[i++] = 6; // var_*

    if (n_in >= 4 + 77) {
        for (int k = 0; k < 77; ++k) leaf[k] = (const float*)d_in[4 + k];
    } else {
        const float* base = (const float*)d_in[4];
        size_t off = 0;
        for (int k = 0; k < 77; ++k) { leaf[k] = base + off; off += (size_t)sz[k]; }
    }
}

static ConvP conv_at(const float* const leaf[77], int b)
{
    ConvP p;
    p.feW = leaf[b + 0];  p.ffW = leaf[b + 1];  p.ffb = leaf[b + 2];
    p.ff_lnb = leaf[b + 3]; p.ff_lng = leaf[b + 4];
    p.flW = leaf[b + 5];  p.flb = leaf[b + 6];  p.frW = leaf[b + 7];
    p.o1W = leaf[b + 8];  p.o1b = leaf[b + 9];  p.o2W = leaf[b + 10]; p.o2b = leaf[b + 11];
    p.pc_lnb = leaf[b + 12]; p.pc_lng = leaf[b + 13];
    return p;
}

static void run_gemm(hipStream_t s, const float* A, const float* B, const float* bias,
                     float* C, int M, int N, int K, int ldc, int relu)
{
    int tiles = ((M + 15) / 16) * ((N + 63) / 64);   // 16x64 strips
    int blocks = (tiles + 7) / 8;                    // 8 waves (strips) per 256-thread block
    if (N == 64 && K == 64)
        gemm16_wmma_f32<64, 64><<<blocks, 256, 0, s>>>(A, B, bias, C, M, ldc, relu);
    else if (N == 64 && K == 128)
        gemm16_wmma_f32<64, 128><<<blocks, 256, 0, s>>>(A, B, bias, C, M, ldc, relu);
    else if (N == 128 && K == 128)
        gemm16_wmma_f32<128, 128><<<blocks, 256, 0, s>>>(A, B, bias, C, M, ldc, relu);
    else if (N == 3 && K == 128)
        gemm16_wmma_f32<3, 128><<<blocks, 256, 0, s>>>(A, B, bias, C, M, ldc, relu);
}

static void run_bgc(hipStream_t s, const ConvP& P,
                    const float* left, int nsrc, const int* src,
                    const float* right, int ntgt, const int* tgt,
                    const float* deg, float* outBuf, int E, const float* efn,
                    float* Tsrc, float* Ttgt, float* Sbuf, float* C128)
{
    // feature_module_right (no bias) on source nodes; feature_module_left on targets
    run_gemm(s, left,  P.frW, nullptr, Tsrc, nsrc, 64, 64, 64, 0);
    run_gemm(s, right, P.flW, P.flb,  Ttgt, ntgt, 64, 64, 64, 0);
    hipMemsetAsync(Sbuf, 0, (size_t)ntgt * 64 * sizeof(float), s);
    edge_msg_kernel<<<(E + 7) / 8, 256, 0, s>>>(Tsrc, Ttgt, efn, P.feW,
                                                P.ff_lng, P.ff_lnb, src, tgt, Sbuf, E);
    // hoisted: agg = (Σ relu(LN(msg))) @ ff_W  (+ deg*ff_b folded into LN step)
    run_gemm(s, Sbuf, P.ffW, nullptr, Tsrc /*=agg*/, ntgt, 64, 64, 64, 0);
    ln_concat_kernel<<<(ntgt + 255) / 256, 256, 0, s>>>(Tsrc, deg, P.ffb,
                                                        P.pc_lng, P.pc_lnb, right, C128, ntgt);
    run_gemm(s, C128, P.o1W, P.o1b, Ttgt /*=hmid*/, ntgt, 64, 128, 64, 1);
    run_gemm(s, Ttgt, P.o2W, P.o2b, outBuf, ntgt, 64, 64, 64, 0);
}

extern "C" void kernel_launch(void* const* d_in, const int* in_sizes, int n_in,
                              void* d_out, int out_size, void* d_ws, size_t ws_size,
                              hipStream_t stream)
{
    const float* consF = (const float*)d_in[0];
    const int*   eidx  = (const int*)d_in[1];
    const float* edgeF = (const float*)d_in[2];
    const float* varF  = (const float*)d_in[3];

    const int nc = in_sizes[0] / 4;     // 20000
    const int E  = in_sizes[2];         // 500000
    const int nv = in_sizes[3] / 6;     // 40000
    const int* cIdx = eidx;             // edge_indices[0]
    const int* vIdx = eidx + E;         // edge_indices[1]

    const float* leaf[77];
    build_leaves(d_in, n_in, leaf);

    // ---- workspace carve-up (~84 MB fp32) ----
    float* ws = (float*)d_ws;
    size_t o = 0;
    float* consA = ws + o; o += (size_t)nc * 64;
    float* consB = ws + o; o += (size_t)nc * 64;
    float* varA  = ws + o; o += (size_t)nv * 64;
    float* varB  = ws + o; o += (size_t)nv * 64;
    float* Tsrc  = ws + o; o += (size_t)nv * 64;   // also: embed h1, agg
    float* Ttgt  = ws + o; o += (size_t)nv * 64;   // also: hmid
    float* Sbuf  = ws + o; o += (size_t)nv * 64;   // segment sum; also head hidden (P*128)
    float* C128  = ws + o; o += (size_t)nv * 128;  // concat; also joined
    float* efn   = ws + o; o += (size_t)E;
    float* degc  = ws + o; o += (size_t)nc;
    float* degv  = ws + o; o += (size_t)nv;
    if (ws_size < o * sizeof(float)) return;       // workspace too small: bail

    // ---- invariants: edge LN scalar, degree counts ----
    edge_ln_kernel<<<(E + 255) / 256, 256, 0, stream>>>(edgeF, leaf[35], leaf[34], efn, E);
    hipMemsetAsync(degc, 0, (size_t)nc * sizeof(float), stream);
    hipMemsetAsync(degv, 0, (size_t)nv * sizeof(float), stream);
    deg_kernel<<<(E + 255) / 256, 256, 0, stream>>>(cIdx, degc, E);
    deg_kernel<<<(E + 255) / 256, 256, 0, stream>>>(vIdx, degv, E);

    // ---- embeddings ----
    embed_front_kernel<4><<<(nc + 255) / 256, 256, 0, stream>>>(
        consF, leaf[33], leaf[32], leaf[28], leaf[30], Tsrc, nc);
    run_gemm(stream, Tsrc, leaf[29], leaf[31], consA, nc, 64, 64, 64, 1);
    embed_front_kernel<6><<<(nv + 255) / 256, 256, 0, stream>>>(
        varF, leaf[76], leaf[75], leaf[71], leaf[73], Tsrc, nv);
    run_gemm(stream, Tsrc, leaf[72], leaf[74], varA, nv, 64, 64, 64, 1);

    float* consCur = consA; float* consNxt = consB;
    float* varCur  = varA;  float* varNxt  = varB;

    // ---- 2 message-passing layers ----
    for (int i = 0; i < 2; ++i) {
        ConvP Pv = conv_at(leaf, 43 + 14 * i);  // v2c
        run_bgc(stream, Pv, varCur, nv, vIdx, consCur, nc, cIdx, degc,
                consNxt, E, efn, Tsrc, Ttgt, Sbuf, C128);
        { float* t = consCur; consCur = consNxt; consNxt = t; }

        ConvP Pc = conv_at(leaf, 0 + 14 * i);   // c2v
        run_bgc(stream, Pc, consCur, nc, cIdx, varCur, nv, vIdx, degv,
                varNxt, E, efn, Tsrc, Ttgt, Sbuf, C128);
        { float* t = varCur; varCur = varNxt; varNxt = t; }

        literals_kernel<<<((size_t)nv * 64 + 255) / 256, 256, 0, stream>>>(
            varCur, leaf[36 + 2 * i], leaf[37 + 2 * i], varNxt, nv);
        { float* t = varCur; varCur = varNxt; varNxt = t; }
    }

    // ---- output head ----
    const int P = nv / 2;                        // 20000 literal pairs
    join_kernel<<<((size_t)P * 128 + 255) / 256, 256, 0, stream>>>(varCur, C128, P);
    run_gemm(stream, C128, leaf[40], leaf[42], Sbuf, P, 128, 128, 128, 1);
    run_gemm(stream, Sbuf, leaf[41], nullptr, (float*)d_out, P, 3, 128, 3, 0);
}